// SelectiveSSM_82411832475828
// MI455X (gfx1250) — compile-verified
//
#include <hip/hip_runtime.h>
#include <hip/hip_bf16.h>

// ---------------------------------------------------------------------------
// Selective SSM (Mamba-style) forward for MI455X / gfx1250.
//   BATCH=4, SEQ=4096, D_MODEL=1024, D_STATE=16, DT_RANK=64
// Pipeline:
//   K1: xp = x @ W_xproj^T   (WMMA f32 16x16x4; W_xproj staged via TDM
//       tensor_load_to_lds, double-buffered, s_wait_tensorcnt sync;
//       TDM pad feature gives a 68-float LDS row stride -> bank-conflict-free)
//   K2: delta = softplus(xp[:, :64] @ W_dt^T + b_dt)   (WMMA f32 16x16x4)
//   K3: chunked scan pass 1: per-chunk (prod dA, local h)   [parallel]
//   K4: chunk combine: exact h at each chunk start           [64-step scan]
//   K5: chunked scan pass 2: replay chunk from exact h, emit y (+ x*D)
// ---------------------------------------------------------------------------

typedef float v2f __attribute__((ext_vector_type(2)));
typedef float v8f __attribute__((ext_vector_type(8)));
typedef unsigned int v4u __attribute__((ext_vector_type(4)));
typedef int v4i __attribute__((ext_vector_type(4)));
typedef int v8i __attribute__((ext_vector_type(8)));

#define D_MODEL 1024
#define D_STATE 16
#define DT_RANK 64
#define BATCH   4
#define SEQ     4096
#define NTOK    (BATCH * SEQ)     // 16384
#define XP_COLS 96                // DT_RANK + 2*D_STATE
#define NCHUNK  64
#define CHLEN   64                // SEQ / NCHUNK

#define KB      64                // K-slice staged per TDM transfer
#define BSTRIDE 68                // 64 data DWORDs + 4 pad DWORDs per row

// ---------------------------------------------------------------------------
// K1: xp[t, 0:96] = sum_k x[t,k] * W_xproj[n,k]
// One wave per 16-token M-tile (8 tiles/block). A comes from global (x is
// streamed exactly once). B (W_xproj) is staged into LDS by the Tensor Data
// Mover: one 96x64 tile (24 KB) per K-slice, double buffered, padded to a
// 68-float row stride by the TDM pad feature so the 16-lane column reads are
// bank-conflict-free ((4*row + col) mod 64 covers 32 distinct banks/wave).
// WMMA f32 16x16x4 operand layout (wave32):
//   A (16x4): lane L<16 holds M=L, K={k0,k0+1}; lane L+16 holds K={k0+2,k0+3}
//   B (4x16): lane L<16 holds N=L, K={k0,k0+1}; high half K={k0+2,k0+3}
//   C/D    : vgpr i -> M = i + 8*(lane>=16), N = lane&15
// ---------------------------------------------------------------------------
__global__ __launch_bounds__(256) void k1_xproj(
    const float* __restrict__ x, const float* __restrict__ Wx,
    float* __restrict__ xp)
{
  __shared__ float sB[2][XP_COLS * BSTRIDE];   // 2 x 26112 B

  const int wid  = threadIdx.x >> 5;
  const int lane = threadIdx.x & 31;
  const int mt   = blockIdx.x * 8 + wid;        // 0..1023 M-tiles
  const int m0   = mt * 16;
  const int m    = lane & 15;
  const int hi   = lane >> 4;
  const int col  = lane & 15;

  const float* arow = x + (size_t)(m0 + m) * D_MODEL + 2 * hi;

  // Issue a TDM load of W_xproj[:, k0:k0+64] into sB[buf] (row stride 68).
  // D# per CDNA5 ISA.
  auto tdm_issue = [&](int buf, int k0) {
    const unsigned lds_addr =
        (unsigned)(unsigned long long)(uintptr_t)(&sB[buf][0]);  // low 32 bits = LDS byte offset
    const unsigned long long ga = (unsigned long long)(uintptr_t)(Wx + k0);
    v4u g0;
    g0[0] = 1u;                                        // count=1, user descriptor
    g0[1] = lds_addr;                                  // lds_addr [63:32]
    g0[2] = (unsigned)(ga & 0xffffffffu);              // global_addr [95:64]
    g0[3] = (unsigned)((ga >> 32) & 0x1ffffffu)        // global_addr [120:96]
          | (2u << 30);                                // type=2 ("image")
    v8i g1;
    g1[0] = (2 << 16)      // data_size = 4 bytes
          | (1 << 20)      // pad_enable
          | (5 << 22)      // pad_interval: 64 DWORDs
          | (3 << 25);     // pad_amount:   4 DWORDs  -> stride 68 floats
    g1[1] = (int)((KB & 0xffffu) << 16);               // tensor_dim0 lo16 @ [63:48]
    g1[2] = (int)((KB >> 16) | ((XP_COLS & 0xffffu) << 16)); // dim0 hi / dim1 lo
    g1[3] = (int)((XP_COLS >> 16) | (KB << 16));       // dim1 hi / tile_dim0=64
    g1[4] = (int)(XP_COLS);                            // tile_dim1=96, tile_dim2=0
    g1[5] = (int)D_MODEL;                              // tensor_dim0_stride lo32 = 1024
    g1[6] = 0;                                         // stride0 hi / stride1 lo
    g1[7] = 0;                                         // stride1 hi
    v4i gz4 = {0, 0, 0, 0};                            // 2-D tile: groups 2/3 unused
    v8i gz8 = {0, 0, 0, 0, 0, 0, 0, 0};
    __builtin_amdgcn_tensor_load_to_lds(g0, g1, gz4, gz4, gz8, 0);
  };

  v8f acc[6] = {};

  if (threadIdx.x == 0) tdm_issue(0, 0);               // prologue (EXEC!=0 lane issues)

  for (int blk = 0; blk < D_MODEL / KB; ++blk) {
    if (threadIdx.x == 0) __builtin_amdgcn_s_wait_tensorcnt(0);
    __syncthreads();   // slice `blk` visible; compute on other buffer finished
    if (threadIdx.x == 0 && blk + 1 < D_MODEL / KB)
      tdm_issue((blk + 1) & 1, (blk + 1) * KB);

    const float* bbuf = &sB[blk & 1][0];
    const int k0 = blk * KB;
#pragma unroll
    for (int kk = 0; kk < KB; kk += 4) {
      v2f a = *(const v2f*)(arow + k0 + kk);
#pragma unroll
      for (int j = 0; j < 6; ++j) {
        v2f b = *(const v2f*)(bbuf + (j * 16 + col) * BSTRIDE + kk + 2 * hi);
        acc[j] = __builtin_amdgcn_wmma_f32_16x16x4_f32(
            false, a, false, b, (short)0, acc[j], false, false);
      }
    }
    __syncthreads();   // all waves done with sB[blk&1] before it is reloaded
  }

#pragma unroll
  for (int j = 0; j < 6; ++j)
#pragma unroll
    for (int i = 0; i < 8; ++i) {
      const int row = m0 + i + 8 * hi;
      xp[(size_t)row * XP_COLS + j * 16 + col] = acc[j][i];
    }
}

// ---------------------------------------------------------------------------
// K2: delta[t, d] = softplus( sum_r xp[t,r] * W_dt[d,r] + b_dt[d] )
// One wave per 16x16 tile, K = 64 (16 wmma steps, fully unrolled).
// W_dt is 256 KB total -> L2 resident; no staging needed.
// ---------------------------------------------------------------------------
__global__ __launch_bounds__(256) void k2_delta(
    const float* __restrict__ xp, const float* __restrict__ Wdt,
    const float* __restrict__ bdt, float* __restrict__ delta)
{
  const int wid  = threadIdx.x >> 5;
  const int lane = threadIdx.x & 31;
  const int tile = blockIdx.x * 8 + wid;        // 0..65535
  const int nt   = tile & 63;
  const int mt   = tile >> 6;
  const int m0   = mt * 16, n0 = nt * 16;
  const int m    = lane & 15;
  const int hi   = lane >> 4;
  const int col  = lane & 15;

  const float* arow = xp  + (size_t)(m0 + m) * XP_COLS + 2 * hi;
  const float* brow = Wdt + (size_t)(n0 + col) * DT_RANK + 2 * hi;

  v8f acc = {};
#pragma unroll
  for (int k0 = 0; k0 < DT_RANK; k0 += 4) {
    v2f a = *(const v2f*)(arow + k0);
    v2f b = *(const v2f*)(brow + k0);
    acc = __builtin_amdgcn_wmma_f32_16x16x4_f32(
        false, a, false, b, (short)0, acc, false, false);
  }

  const float bias = bdt[n0 + col];
#pragma unroll
  for (int i = 0; i < 8; ++i) {
    const int row = m0 + i + 8 * hi;
    const float v  = acc[i] + bias;
    const float sp = (v > 20.f) ? v : log1pf(__expf(v));  // stable softplus
    delta[(size_t)row * D_MODEL + n0 + col] = sp;
  }
}

// ---------------------------------------------------------------------------
// K3: scan pass 1. Thread = (b, d, chunk, n). Over 64 steps:
//   dA = exp(dt*A_n); h = h*dA + x*dt*B_n; ap *= dA
// No cross-lane communication; fully parallel (4.2M threads).
// ---------------------------------------------------------------------------
__global__ __launch_bounds__(256) void k3_scan1(
    const float* __restrict__ x, const float* __restrict__ xp,
    const float* __restrict__ delta, const float* __restrict__ A_log,
    float* __restrict__ aprod, float* __restrict__ hloc)
{
  const unsigned gid = blockIdx.x * 256u + threadIdx.x;
  const int n  = gid & 15;
  const unsigned u  = gid >> 4;
  const int ch = u & (NCHUNK - 1);
  const unsigned cd = u >> 6;          // b*D_MODEL + d
  const int d = cd & (D_MODEL - 1);
  const int b = cd >> 10;

  const float A = -__expf(A_log[d * D_STATE + n]);
  const int tok0 = b * SEQ + ch * CHLEN;
  const float* dtp = delta + (size_t)tok0 * D_MODEL + d;
  const float* xvp = x     + (size_t)tok0 * D_MODEL + d;
  const float* Bp  = xp    + (size_t)tok0 * XP_COLS + DT_RANK + n;

  float h = 0.f, ap = 1.f;
  for (int s = 0; s < CHLEN; ++s) {
    __builtin_prefetch(dtp + (size_t)(s + 16) * D_MODEL, 0, 1);
    __builtin_prefetch(xvp + (size_t)(s + 16) * D_MODEL, 0, 1);
    const float dt = dtp[(size_t)s * D_MODEL];
    const float xv = xvp[(size_t)s * D_MODEL];
    const float Bn = Bp[(size_t)s * XP_COLS];
    const float dA = __expf(dt * A);
    h  = fmaf(h, dA, xv * dt * Bn);
    ap *= dA;
  }
  aprod[gid] = ap;
  hloc[gid]  = h;
}

// ---------------------------------------------------------------------------
// K4: combine chunk summaries -> exact state at each chunk start.
// Thread = (b, d, n): 64-step sequential scan over chunks.
// ---------------------------------------------------------------------------
__global__ __launch_bounds__(256) void k4_combine(
    const float* __restrict__ aprod, const float* __restrict__ hloc,
    float* __restrict__ hstart)
{
  const unsigned tid = blockIdx.x * 256u + threadIdx.x;  // 0..65535
  const int n = tid & 15;
  const unsigned cd = tid >> 4;
  float h = 0.f;
  for (int ch = 0; ch < NCHUNK; ++ch) {
    const unsigned idx = (cd * NCHUNK + ch) * 16u + n;
    hstart[idx] = h;
    h = fmaf(h, aprod[idx], hloc[idx]);
  }
}

// ---------------------------------------------------------------------------
// K5: scan pass 2. Replay chunk from exact start state, reduce over the 16
// state lanes via DS shuffles, and write y = sum_n h_n*C_n + x*D.
// ---------------------------------------------------------------------------
__global__ __launch_bounds__(256) void k5_scan2(
    const float* __restrict__ x, const float* __restrict__ xp,
    const float* __restrict__ delta, const float* __restrict__ A_log,
    const float* __restrict__ Dvec, const float* __restrict__ hstart,
    float* __restrict__ y)
{
  const unsigned gid = blockIdx.x * 256u + threadIdx.x;
  const int n  = gid & 15;
  const unsigned u  = gid >> 4;
  const int ch = u & (NCHUNK - 1);
  const unsigned cd = u >> 6;
  const int d = cd & (D_MODEL - 1);
  const int b = cd >> 10;

  const float A  = -__expf(A_log[d * D_STATE + n]);
  const float Dd = Dvec[d];
  const int tok0 = b * SEQ + ch * CHLEN;
  const float* dtp = delta + (size_t)tok0 * D_MODEL + d;
  const float* xvp = x     + (size_t)tok0 * D_MODEL + d;
  const float* Bp  = xp    + (size_t)tok0 * XP_COLS + DT_RANK + n;
  const float* Cp  = xp    + (size_t)tok0 * XP_COLS + DT_RANK + D_STATE + n;
  float* yp = y + (size_t)tok0 * D_MODEL + d;

  float h = hstart[gid];
  for (int s = 0; s < CHLEN; ++s) {
    __builtin_prefetch(dtp + (size_t)(s + 16) * D_MODEL, 0, 1);
    __builtin_prefetch(xvp + (size_t)(s + 16) * D_MODEL, 0, 1);
    const float dt = dtp[(size_t)s * D_MODEL];
    const float xv = xvp[(size_t)s * D_MODEL];
    const float Bn = Bp[(size_t)s * XP_COLS];
    const float Cn = Cp[(size_t)s * XP_COLS];
    const float dA = __expf(dt * A);
    h = fmaf(h, dA, xv * dt * Bn);
    float p = h * Cn;
    p += __shfl_xor(p, 1, 16);
    p += __shfl_xor(p, 2, 16);
    p += __shfl_xor(p, 4, 16);
    p += __shfl_xor(p, 8, 16);
    if (n == 0) yp[(size_t)s * D_MODEL] = fmaf(xv, Dd, p);
  }
}

// ---------------------------------------------------------------------------
extern "C" void kernel_launch(void* const* d_in, const int* in_sizes, int n_in,
                              void* d_out, int out_size, void* d_ws, size_t ws_size,
                              hipStream_t stream)
{
  const float* x      = (const float*)d_in[0];   // (4,4096,1024)
  const float* Wxproj = (const float*)d_in[1];   // (96,1024)
  const float* Wdt    = (const float*)d_in[2];   // (1024,64)
  const float* bdt    = (const float*)d_in[3];   // (1024,)
  const float* A_log  = (const float*)d_in[4];   // (1024,16)
  const float* Dvec   = (const float*)d_in[5];   // (1024,)
  float* y = (float*)d_out;

  // workspace layout (floats)
  float* ws = (float*)d_ws;
  const size_t XP_ELEMS    = (size_t)NTOK * XP_COLS;                 // 1.5M
  const size_t DELTA_ELEMS = (size_t)NTOK * D_MODEL;                 // 16.8M
  const size_t SUMM_ELEMS  = (size_t)BATCH * D_MODEL * NCHUNK * 16;  // 4.2M
  float* xp     = ws;
  float* delta  = xp + XP_ELEMS;
  float* aprod  = delta + DELTA_ELEMS;
  float* hloc   = aprod + SUMM_ELEMS;
  float* hstart = hloc + SUMM_ELEMS;

  // K1: 1024 M-tiles, 8 waves/block
  k1_xproj<<<128, 256, 0, stream>>>(x, Wxproj, xp);
  // K2: 1024 x 64 tiles, 8 waves/block
  k2_delta<<<8192, 256, 0, stream>>>(xp, Wdt, bdt, delta);
  // K3: 4*1024*64*16 = 4,194,304 threads
  k3_scan1<<<16384, 256, 0, stream>>>(x, xp, delta, A_log, aprod, hloc);
  // K4: 4*1024*16 = 65,536 threads
  k4_combine<<<256, 256, 0, stream>>>(aprod, hloc, hstart);
  // K5: 4,194,304 threads
  k5_scan2<<<16384, 256, 0, stream>>>(x, xp, delta, A_log, Dvec, hstart, y);
}